// NeuronCircuit_42271068127541
// MI455X (gfx1250) — compile-verified
//
#include <hip/hip_runtime.h>
#include <hip/hip_bf16.h>

// ---------------------------------------------------------------------------
// NeuronCircuit pipeline for MI455X (gfx1250), fp32 end-to-end.
// GEMMs + attention matmuls: V_WMMA_F32_16X16X4_F32 (fp32-exact).
// GEMM tile staging: Tensor Data Mover (tensor_load_to_lds, TENSORcnt).
// ---------------------------------------------------------------------------

#ifndef USE_TDM
#define USE_TDM 1
#endif

typedef float v2f __attribute__((ext_vector_type(2)));
typedef float v8f __attribute__((ext_vector_type(8)));
typedef unsigned int uint32x4 __attribute__((ext_vector_type(4)));
typedef int int32x4 __attribute__((ext_vector_type(4)));
typedef int int32x8 __attribute__((ext_vector_type(8)));

static __device__ __forceinline__ v8f wmma_f32(v2f a, v2f b, v8f c) {
  return __builtin_amdgcn_wmma_f32_16x16x4_f32(false, a, false, b, (short)0, c,
                                               false, false);
}

static constexpr int Bb = 8, Ss = 1024, Dd = 1024, Hh = 16, DH = 64, Rr = 256;

// ------------------------- TDM descriptor helpers --------------------------
// D# group1 (ISA cdna5 §8.4): data_size=4B, pad enabled, 2D tile.
//   tile_w elements per row (contiguous), tile_h rows, row stride in elements.
//   pad_i: code, interval = 2^(pad_i+1) dwords; pad_a: code, amount = pad_a+1 dw.
static __device__ __forceinline__ int32x8 tdm_g1(unsigned tile_w, unsigned tile_h,
                                                 unsigned stride, unsigned pad_i,
                                                 unsigned pad_a) {
  union { unsigned long long q[4]; int32x8 v; } u;
  u.q[0] = (unsigned long long)((2u << 16) | (1u << 20) | (pad_i << 22) |
                                (pad_a << 25))                    // flags
           | ((unsigned long long)(tile_w & 0xFFFFu) << 48);      // tensor_dim0 lo
  u.q[1] = (unsigned long long)(tile_w >> 16)                     // tensor_dim0 hi
           | ((unsigned long long)tile_h << 16)                   // tensor_dim1
           | ((unsigned long long)(tile_w & 0xFFFFu) << 48);      // tile_dim0
  u.q[2] = (unsigned long long)(tile_h & 0xFFFFu)                 // tile_dim1
           | ((unsigned long long)stride << 32);                  // dim0_stride lo32
  u.q[3] = 0ull;                                                  // stride hi16=0
  return u.v;
}

static __device__ __forceinline__ void tdm_load_2d(unsigned lds_addr,
                                                   const float* g, int32x8 g1) {
  unsigned long long ga = (unsigned long long)(uintptr_t)g;
  uint32x4 g0;
  g0.x = 1u;                                   // count=1 (valid user D#)
  g0.y = lds_addr;                             // LDS byte address
  g0.z = (unsigned)ga;                         // global_addr[31:0]
  g0.w = (unsigned)(ga >> 32) | (2u << 30);    // global_addr[56:32] | type=2
  int32x4 z4 = {0, 0, 0, 0};
#if defined(__clang_major__) && __clang_major__ >= 23
  int32x8 z8 = {0, 0, 0, 0, 0, 0, 0, 0};
  __builtin_amdgcn_tensor_load_to_lds(g0, g1, z4, z4, z8, 0);
#else
  __builtin_amdgcn_tensor_load_to_lds(g0, g1, z4, z4, 0);
#endif
}

static __device__ __forceinline__ unsigned lds_off(const void* p) {
  return (unsigned)(uintptr_t)p;   // generic LDS address: low 32 bits = offset
}

// ------------------------------- SSM stage ---------------------------------

__global__ __launch_bounds__(64) void k_xb(const float* __restrict__ x,
                                           const float* __restrict__ Bm,
                                           float* __restrict__ xb) {
  __shared__ float xr[1024];
  int bs = blockIdx.x;          // b*1024 + s
  int b = bs >> 10, s = bs & 1023;
  const float* xp = x + (long)bs * Dd;
  for (int i = threadIdx.x; i < 256; i += 64)
    ((float4*)xr)[i] = ((const float4*)xp)[i];
  __syncthreads();
  int n = threadIdx.x;
  float acc = 0.f;
  for (int d = 0; d < Dd; ++d) acc += xr[d] * Bm[d * 64 + n];
  xb[((long)s * Bb + b) * 64 + n] = acc;
}

__global__ __launch_bounds__(512) void k_scan(const float* __restrict__ xb,
                                              const float* __restrict__ A,
                                              const float* __restrict__ Wimp,
                                              float* __restrict__ hproj) {
  __shared__ float h[Bb][64];
  int t = threadIdx.x;
  int b = t >> 6, n = t & 63;
  h[b][n] = 0.f;
  __syncthreads();
  for (int s = 0; s < Ss; ++s) {
    float acc = xb[((long)s * Bb + b) * 64 + n];
    for (int m = 0; m < 64; ++m) acc += h[b][m] * A[m * 64 + n];
    __syncthreads();
    h[b][n] = acc;
    __syncthreads();
  }
  for (int i = 0; i < 16; ++i) {
    int o = i * 512 + t;
    int ob = o >> 10, d = o & 1023;
    float acc = 0.f;
    for (int m = 0; m < 64; ++m) acc += h[ob][m] * Wimp[m * Dd + d];
    hproj[o] = acc;
  }
}

__global__ __launch_bounds__(256) void k_implog(const float* __restrict__ x,
                                                const float* __restrict__ hp,
                                                float* __restrict__ lg) {
  int wid = blockIdx.x * 8 + (threadIdx.x >> 5);
  int lane = threadIdx.x & 31;
  int b = wid >> 10;
  const float* xp = x + (long)wid * Dd;
  const float* hb = hp + (long)b * Dd;
  float acc = 0.f;
  for (int i = 0; i < 32; ++i) acc += xp[lane + 32 * i] * hb[lane + 32 * i];
  for (int off = 16; off; off >>= 1) acc += __shfl_xor(acc, off, 32);
  if (lane == 0) lg[wid] = acc;
}

__global__ __launch_bounds__(256) void k_softmax_s(const float* __restrict__ lg,
                                                   float* __restrict__ imp) {
  __shared__ float red[256];
  int b = blockIdx.x, t = threadIdx.x;
  const float* lp = lg + (long)b * Ss;
  float mx = -1e30f;
  for (int i = t; i < Ss; i += 256) mx = fmaxf(mx, lp[i]);
  red[t] = mx; __syncthreads();
  for (int off = 128; off; off >>= 1) {
    if (t < off) red[t] = fmaxf(red[t], red[t + off]);
    __syncthreads();
  }
  mx = red[0]; __syncthreads();
  float sum = 0.f;
  for (int i = t; i < Ss; i += 256) sum += __expf(lp[i] - mx);
  red[t] = sum; __syncthreads();
  for (int off = 128; off; off >>= 1) {
    if (t < off) red[t] += red[t + off];
    __syncthreads();
  }
  float inv = 1.f / red[0];
  for (int i = t; i < Ss; i += 256) imp[b * Ss + i] = __expf(lp[i] - mx) * inv;
}

// ------------------------------- routing -----------------------------------

__global__ __launch_bounds__(128) void k_route(const float* __restrict__ x,
                                               const float* __restrict__ Wc,
                                               const float* __restrict__ Wq,
                                               const float* __restrict__ Wk,
                                               const float* __restrict__ Wv,
                                               const float* __restrict__ Wo,
                                               const float* __restrict__ imp,
                                               float* __restrict__ contrib) {
  __shared__ float xr[1024];
  __shared__ float lg[80];
  int bs = blockIdx.x;
  const float* xp = x + (long)bs * Dd;
  int t = threadIdx.x;
  int g = 0, n = 0, Ng = 16;
  const float* W = nullptr;
  if (t < 16)      { g = 0; n = t;      W = Wc; }
  else if (t < 32) { g = 1; n = t - 16; W = Wq; }
  else if (t < 48) { g = 2; n = t - 32; W = Wk; }
  else if (t < 64) { g = 3; n = t - 48; W = Wv; }
  else if (t < 76) { g = 4; n = t - 64; Ng = 12; W = Wo; }
  for (int i = t; i < 256; i += 128)
    ((float4*)xr)[i] = ((const float4*)xp)[i];
  __syncthreads();
  if (t < 76) {
    float acc = 0.f;
    for (int d = 0; d < Dd; ++d) acc += xr[d] * W[d * Ng + n];
    lg[g * 16 + n] = acc;
  }
  __syncthreads();
  if (t < 76) {
    int gs = g * 16;
    float mx = -1e30f;
    for (int i = 0; i < Ng; ++i) mx = fmaxf(mx, lg[gs + i]);
    float sum = 0.f;
    for (int i = 0; i < Ng; ++i) sum += __expf(lg[gs + i] - mx);
    float pref = __expf(lg[gs + n] - mx) / sum;
    contrib[(long)bs * 80 + gs + n] = imp[bs] * pref;
  } else if (t < 80) {
    contrib[(long)bs * 80 + t] = 0.f;
  }
}

__global__ __launch_bounds__(256) void k_wred(const float* __restrict__ contrib,
                                              float* __restrict__ wacc) {
  __shared__ float red[256];
  int b = blockIdx.x / 80, j = blockIdx.x % 80;
  float acc = 0.f;
  for (int s = threadIdx.x; s < Ss; s += 256)
    acc += contrib[((long)b * Ss + s) * 80 + j];
  red[threadIdx.x] = acc; __syncthreads();
  for (int off = 128; off; off >>= 1) {
    if (threadIdx.x < off) red[threadIdx.x] += red[threadIdx.x + off];
    __syncthreads();
  }
  if (threadIdx.x == 0) wacc[b * 80 + j] = red[0];
}

__global__ __launch_bounds__(64) void k_norm(float* __restrict__ wacc) {
  int t = threadIdx.x;
  if (t >= 40) return;
  int b = t / 5, g = t % 5;
  int Ng = (g == 4) ? 12 : 16;
  float* wv = wacc + b * 80 + g * 16;
  float sum = 0.f;
  for (int n = 0; n < Ng; ++n) sum += wv[n];
  float inv = 1.f / (sum + 1e-8f);
  for (int n = 0; n < Ng; ++n) wv[n] *= inv;
}

__global__ void k_mix(float* __restrict__ dst, const float* __restrict__ pool,
                      const float* __restrict__ wacc, int g, int npool, long E) {
  long idx = (long)blockIdx.x * blockDim.x + threadIdx.x;
  if (idx >= 8 * E) return;
  int b = (int)(idx / E);
  long e = idx % E;
  const float* wv = wacc + b * 80 + g * 16;
  float acc = 0.f;
  for (int n = 0; n < npool; ++n) acc += wv[n] * pool[(long)n * E + e];
  dst[idx] = acc;
}

// ------------------------- batched fp32 WMMA GEMM --------------------------
// C[b] (MxN) = A[b] (MxK, lda=K) * B[b] (KxN, ldb=N). 64x64 block tile,
// K-slab 32. Tiles staged by the Tensor Data Mover into padded LDS rows:
//   A: 64 rows x 32dw + 4dw pad  -> As[64][36]
//   B: 32 rows x 64dw + 4dw pad  -> Bs[32][68]
__global__ __launch_bounds__(128) void k_gemm(const float* __restrict__ A,
                                              const float* __restrict__ B,
                                              float* __restrict__ C,
                                              int M, int N, int K,
                                              long sA, long sB, long sC) {
  __shared__ float As[64][36];
  __shared__ float Bs[32][68];
  const int bm = blockIdx.x * 64;
  const int bn = blockIdx.y * 64;
  A += (long)blockIdx.z * sA;
  B += (long)blockIdx.z * sB;
  C += (long)blockIdx.z * sC;
  const int t = threadIdx.x;
  const int w = t >> 5, lane = t & 31;
  const int lhalf = lane >> 4, l16 = lane & 15;
  v8f acc[4];
  for (int j = 0; j < 4; ++j)
    for (int v = 0; v < 8; ++v) acc[j][v] = 0.f;

#if USE_TDM
  const int32x8 gA1 = tdm_g1(/*tile_w=*/32, /*tile_h=*/64, (unsigned)K,
                             /*pad_i: 32dw=*/4u, /*pad_a: 4dw=*/3u);
  const int32x8 gB1 = tdm_g1(/*tile_w=*/64, /*tile_h=*/32, (unsigned)N,
                             /*pad_i: 64dw=*/5u, /*pad_a: 4dw=*/3u);
  const unsigned ldsA = lds_off(&As[0][0]);
  const unsigned ldsB = lds_off(&Bs[0][0]);
#endif

  for (int k0 = 0; k0 < K; k0 += 32) {
#if USE_TDM
    if (t < 32) {  // one DMA per workgroup (TDM ignores EXEC; wave-uniform)
      tdm_load_2d(ldsA, A + (long)bm * K + k0, gA1);
      tdm_load_2d(ldsB, B + (long)k0 * N + bn, gB1);
      __builtin_amdgcn_s_wait_tensorcnt((short)0);
    }
    __syncthreads();
#else
    {
      int row = t >> 1, half = (t & 1) * 16;
      const float* src = A + (long)(bm + row) * K + k0 + half;
      float* dst = &As[row][half];
#pragma unroll
      for (int j = 0; j < 4; ++j)
        *(float4*)(dst + 4 * j) = *(const float4*)(src + 4 * j);
    }
    {
      int row = t >> 2, q = (t & 3) * 16;
      const float* src = B + (long)(k0 + row) * N + bn + q;
      float* dst = &Bs[row][q];
#pragma unroll
      for (int j = 0; j < 4; ++j)
        *(float4*)(dst + 4 * j) = *(const float4*)(src + 4 * j);
    }
    __syncthreads();
#endif
#pragma unroll
    for (int kk = 0; kk < 8; ++kk) {
      int kb = kk * 4 + lhalf * 2;
      v2f a = *(const v2f*)&As[w * 16 + l16][kb];
#pragma unroll
      for (int j = 0; j < 4; ++j) {
        v2f bf;
        bf.x = Bs[kb][j * 16 + l16];
        bf.y = Bs[kb + 1][j * 16 + l16];
        acc[j] = wmma_f32(a, bf, acc[j]);
      }
    }
    __syncthreads();
  }
#pragma unroll
  for (int j = 0; j < 4; ++j)
#pragma unroll
    for (int v = 0; v < 8; ++v) {
      int row = bm + w * 16 + v + lhalf * 8;
      int col = bn + j * 16 + l16;
      C[(long)row * N + col] = acc[j][v];
    }
}

// --------------------------- flash attention -------------------------------
__global__ __launch_bounds__(128) void k_attn(const float* __restrict__ Q,
                                              const float* __restrict__ Km,
                                              const float* __restrict__ V,
                                              float* __restrict__ Out) {
  __shared__ float P[4][16][68];
  __shared__ float mrow[4][16], lrow[4][16], frow[4][16];
  const int qt = blockIdx.x, hh = blockIdx.y, b = blockIdx.z;
  const int t = threadIdx.x, w = t >> 5, lane = t & 31;
  const int lhalf = lane >> 4, l16 = lane & 15;
  const int q0 = qt * 64 + w * 16;
  const long baseQ = ((long)b * Ss + q0) * Dd + hh * DH;
  const long baseKV = (long)b * Ss * Dd + hh * DH;

  v8f o[4];
  for (int j = 0; j < 4; ++j)
    for (int v = 0; v < 8; ++v) o[j][v] = 0.f;
  if (lane < 16) { mrow[w][lane] = -1e30f; lrow[w][lane] = 0.f; }

  for (int kt = 0; kt <= qt; ++kt) {
    const int k0 = kt * 64;
    v8f s[4];
    for (int j = 0; j < 4; ++j)
      for (int v = 0; v < 8; ++v) s[j][v] = 0.f;
#pragma unroll 4
    for (int kk = 0; kk < 16; ++kk) {
      int kb = kk * 4 + lhalf * 2;
      v2f a;
      a.x = Q[baseQ + (long)l16 * Dd + kb];
      a.y = Q[baseQ + (long)l16 * Dd + kb + 1];
#pragma unroll
      for (int j = 0; j < 4; ++j) {
        const float* kp = Km + baseKV + (long)(k0 + j * 16 + l16) * Dd + kb;
        v2f bf; bf.x = kp[0]; bf.y = kp[1];
        s[j] = wmma_f32(a, bf, s[j]);
      }
    }
#pragma unroll
    for (int j = 0; j < 4; ++j)
#pragma unroll
      for (int v = 0; v < 8; ++v) {
        int r = v + lhalf * 8;
        int kidx = k0 + j * 16 + l16;
        P[w][r][j * 16 + l16] =
            (kidx <= q0 + r) ? s[j][v] * 0.125f : -1e30f;
      }
    if (lane < 16) {
      float m_old = mrow[w][lane];
      float tmax = -1e30f;
      for (int c = 0; c < 64; ++c) tmax = fmaxf(tmax, P[w][lane][c]);
      float m_new = fmaxf(m_old, tmax);
      float fac = __expf(m_old - m_new);
      float ssum = 0.f;
      for (int c = 0; c < 64; ++c) {
        float p = __expf(P[w][lane][c] - m_new);
        P[w][lane][c] = p;
        ssum += p;
      }
      mrow[w][lane] = m_new;
      lrow[w][lane] = lrow[w][lane] * fac + ssum;
      frow[w][lane] = fac;
    }
#pragma unroll
    for (int v = 0; v < 8; ++v) {
      float f = frow[w][v + lhalf * 8];
      for (int j = 0; j < 4; ++j) o[j][v] *= f;
    }
#pragma unroll 4
    for (int kk = 0; kk < 16; ++kk) {
      int kb = kk * 4 + lhalf * 2;
      v2f a = *(const v2f*)&P[w][l16][kb];
#pragma unroll
      for (int j = 0; j < 4; ++j) {
        const float* vp = V + baseKV + (long)(k0 + kb) * Dd + j * 16 + l16;
        v2f bf; bf.x = vp[0]; bf.y = vp[Dd];
        o[j] = wmma_f32(a, bf, o[j]);
      }
    }
  }
#pragma unroll
  for (int v = 0; v < 8; ++v) {
    float inv = 1.f / lrow[w][v + lhalf * 8];
    int r = q0 + v + lhalf * 8;
    for (int j = 0; j < 4; ++j)
      Out[((long)b * Ss + r) * Dd + hh * DH + j * 16 + l16] = o[j][v] * inv;
  }
}

// ------------------------------ launcher -----------------------------------

extern "C" void kernel_launch(void* const* d_in, const int* in_sizes, int n_in,
                              void* d_out, int out_size, void* d_ws,
                              size_t ws_size, hipStream_t stream) {
  const float* x     = (const float*)d_in[0];
  // d_in[1] = mask (causal tril) -- applied analytically
  const float* A     = (const float*)d_in[2];
  const float* Bm    = (const float*)d_in[3];
  const float* Wimp  = (const float*)d_in[4];
  const float* Wc    = (const float*)d_in[5];
  const float* Wq    = (const float*)d_in[6];
  const float* Wk    = (const float*)d_in[7];
  const float* Wv    = (const float*)d_in[8];
  const float* Wo    = (const float*)d_in[9];
  const float* comp  = (const float*)d_in[10];
  const float* expd  = (const float*)d_in[11];
  const float* Opool = (const float*)d_in[12];
  float* out = (float*)d_out;

  float* p = (float*)d_ws;
  float* xb      = p; p += (long)Ss * Bb * 64;
  float* hproj   = p; p += Bb * Dd;
  float* logits  = p; p += Bb * Ss;
  float* imp     = p; p += Bb * Ss;
  float* contrib = p; p += (long)Bb * Ss * 80;
  float* wacc    = p; p += Bb * 80;
  float* Pc      = p; p += (long)Bb * Dd * Rr;
  float* Eq      = p; p += (long)Bb * Rr * Dd;
  float* Ek      = p; p += (long)Bb * Rr * Dd;
  float* Ev      = p; p += (long)Bb * Rr * Dd;
  float* hb      = p; p += (long)Bb * Ss * Rr;
  float* Qb      = p; p += (long)Bb * Ss * Dd;
  float* Kb      = p; p += (long)Bb * Ss * Dd;
  float* Vb      = p; p += (long)Bb * Ss * Dd;
  float* Osh     = p; p += (long)Bb * Dd * Dd;
  float* attn    = p; p += (long)Bb * Ss * Dd;

  k_xb<<<Bb * Ss, 64, 0, stream>>>(x, Bm, xb);
  k_scan<<<1, 512, 0, stream>>>(xb, A, Wimp, hproj);
  k_implog<<<Bb * Ss / 8, 256, 0, stream>>>(x, hproj, logits);
  k_softmax_s<<<Bb, 256, 0, stream>>>(logits, imp);
  k_route<<<Bb * Ss, 128, 0, stream>>>(x, Wc, Wq, Wk, Wv, Wo, imp, contrib);
  k_wred<<<Bb * 80, 256, 0, stream>>>(contrib, wacc);
  k_norm<<<1, 64, 0, stream>>>(wacc);
  {
    long E = (long)Dd * Rr;
    int blks = (int)((Bb * E + 255) / 256);
    k_mix<<<blks, 256, 0, stream>>>(Pc, comp, wacc, 0, 16, E);
    k_mix<<<blks, 256, 0, stream>>>(Eq, expd, wacc, 1, 16, E);
    k_mix<<<blks, 256, 0, stream>>>(Ek, expd, wacc, 2, 16, E);
    k_mix<<<blks, 256, 0, stream>>>(Ev, expd, wacc, 3, 16, E);
    long EO = (long)Dd * Dd;
    k_mix<<<(int)((Bb * EO + 255) / 256), 256, 0, stream>>>(Osh, Opool, wacc,
                                                            4, 12, EO);
  }
  k_gemm<<<dim3(16, 4, Bb), 128, 0, stream>>>(x, Pc, hb, Ss, Rr, Dd,
                                              (long)Ss * Dd, (long)Dd * Rr,
                                              (long)Ss * Rr);
  k_gemm<<<dim3(16, 16, Bb), 128, 0, stream>>>(hb, Eq, Qb, Ss, Dd, Rr,
                                               (long)Ss * Rr, (long)Rr * Dd,
                                               (long)Ss * Dd);
  k_gemm<<<dim3(16, 16, Bb), 128, 0, stream>>>(hb, Ek, Kb, Ss, Dd, Rr,
                                               (long)Ss * Rr, (long)Rr * Dd,
                                               (long)Ss * Dd);
  k_gemm<<<dim3(16, 16, Bb), 128, 0, stream>>>(hb, Ev, Vb, Ss, Dd, Rr,
                                               (long)Ss * Rr, (long)Rr * Dd,
                                               (long)Ss * Dd);
  k_attn<<<dim3(16, Hh, Bb), 128, 0, stream>>>(Qb, Kb, Vb, attn);
  k_gemm<<<dim3(16, 16, Bb), 128, 0, stream>>>(attn, Osh, out, Ss, Dd, Dd,
                                               (long)Ss * Dd, (long)Dd * Dd,
                                               (long)Ss * Dd);
}